// SoftTopK_14551349199340
// MI455X (gfx1250) — compile-verified
//
#include <hip/hip_runtime.h>
#include <hip/hip_bf16.h>
#include <stdint.h>

// Problem constants (match the reference)
#define SB 8
#define SN 1024
#define SM 256
#define SS 8
#define SK 16
#define SIGMA 1e-4f

typedef float v4f __attribute__((ext_vector_type(4)));

__global__ __launch_bounds__(256) void
soft_topk_kernel(const float* __restrict__ x,
                 const float* __restrict__ noise,
                 float* __restrict__ out) {
    __shared__ float sX[SM];            // shared x row (1 KB)
    __shared__ float tile[SK * SM];     // output tile (16 KB)

    const int g    = blockIdx.x;        // g = b*N + n
    const int tid  = threadIdx.x;
    const int wave = tid >> 5;          // sample index s (wave32)
    const int lane = tid & 31;

    // ---- zero-fill the 16x256 accumulation tile (ds_store_b128) ----
    v4f* tile4 = (v4f*)tile;
    {
        v4f z = {0.f, 0.f, 0.f, 0.f};
#pragma unroll
        for (int c = 0; c < 4; ++c) tile4[c * 256 + tid] = z;
    }

    // ---- wave 0: async-copy x row (256 floats) global -> LDS ----
    // CDNA5 async path: 32 lanes x 16B = 512B per instruction, IOFFSET
    // applies to BOTH the LDS and global address, so two issues cover 1 KB.
    if (wave == 0) {
        uint32_t lds = (uint32_t)(uintptr_t)(&sX[0]) + (uint32_t)(lane * 16);
        uint64_t ga  = (uint64_t)(uintptr_t)(x + (size_t)g * SM) +
                       (uint64_t)(lane * 16);
        asm volatile(
            "global_load_async_to_lds_b128 %0, %1, off\n\t"
            "global_load_async_to_lds_b128 %0, %1, off offset:512\n\t"
            "s_wait_asynccnt 0"
            :: "v"(lds), "v"(ga) : "memory");
    }
    __syncthreads();

    // ---- per-wave sample: build 64-bit order keys for 8 local elements ----
    // key = (monotone(f32) << 8) | element_index  -> u64 argmin == ordered
    // smallest with lowest-index tie-break (matches jax top_k on -x).
    const size_t nbase = (((size_t)g * SS + (size_t)wave) * SM) +
                         (size_t)lane * 8;
    v4f n0 = __builtin_nontemporal_load((const v4f*)(noise + nbase));
    v4f n1 = __builtin_nontemporal_load((const v4f*)(noise + nbase + 4));
    v4f x0 = *(const v4f*)&sX[lane * 8];
    v4f x1 = *(const v4f*)&sX[lane * 8 + 4];

    uint64_t key[8];
#pragma unroll
    for (int j = 0; j < 8; ++j) {
        float nv = (j < 4) ? n0[j] : n1[j - 4];
        float xv = (j < 4) ? x0[j] : x1[j - 4];
        float v  = fmaf(nv, SIGMA, xv);
        uint32_t u = __float_as_uint(v);
        u = (u & 0x80000000u) ? ~u : (u | 0x80000000u);  // monotone map
        key[j] = ((uint64_t)u << 8) | (uint32_t)(lane * 8 + j);
    }

    // ---- 16 x (local min -> wave32 xor-shuffle min -> extract) ----
    for (int k = 0; k < SK; ++k) {
        uint64_t m = key[0];
#pragma unroll
        for (int j = 1; j < 8; ++j) m = (key[j] < m) ? key[j] : m;
#pragma unroll
        for (int off = 16; off > 0; off >>= 1) {
            uint64_t o = (uint64_t)__shfl_xor((unsigned long long)m, off, 32);
            m = (o < m) ? o : m;
        }
        const uint32_t minIdx = (uint32_t)(m & 0xFFu);
        // owner lane retires its element
#pragma unroll
        for (int j = 0; j < 8; ++j)
            if (minIdx == (uint32_t)(lane * 8 + j)) key[j] = ~0ull;
        // accumulate 1/S into the (k, minIdx) cell (ds_add_f32)
        if (lane == 0) atomicAdd(&tile[k * SM + (int)minIdx], 0.125f);
    }
    __syncthreads();

    // ---- coalesced non-temporal streaming store of the 16 KB tile ----
    v4f* out4 = ((v4f*)out) + (size_t)g * (SK * SM / 4);
#pragma unroll
    for (int c = 0; c < 4; ++c) {
        v4f t = tile4[c * 256 + tid];
        __builtin_nontemporal_store(t, out4 + c * 256 + tid);
    }
}

extern "C" void kernel_launch(void* const* d_in, const int* in_sizes, int n_in,
                              void* d_out, int out_size, void* d_ws, size_t ws_size,
                              hipStream_t stream) {
    (void)in_sizes; (void)n_in; (void)out_size; (void)d_ws; (void)ws_size;
    const float* x     = (const float*)d_in[0];   // (8,1024,256)
    const float* noise = (const float*)d_in[1];   // (8,1024,8,256)
    float* out = (float*)d_out;                   // (8,1024,16,256)
    dim3 grid(SB * SN);                           // one workgroup per (b,n)
    dim3 block(256);                              // 8 wave32 = 8 samples
    soft_topk_kernel<<<grid, block, 0, stream>>>(x, noise, out);
}